// GCN_Encoder_6090263626103
// MI455X (gfx1250) — compile-verified
//
#include <hip/hip_runtime.h>
#include <cstddef>

// ---------------------------------------------------------------------------
// GCN encoder for MI455X (gfx1250, wave32, WMMA).
//  Strategy: aggregate-then-project (A·(XW) == (A·X)·W), shared aggregation
//  for the mu/logstd heads, f32 WMMA (16x16x4) for all wide GEMMs.
//  Weights are pre-transposed (WT: N x 64) so both A and B fragments load as
//  single 8-byte global_load_b64 ops: 5 vmem per 4 v_wmma in the inner loop.
// ---------------------------------------------------------------------------

typedef float v2f __attribute__((ext_vector_type(2)));
typedef float v8f __attribute__((ext_vector_type(8)));

#define N_NODES 50000
#define N_EDGES 800000
#define HID 64
#define OUTC 768
#define M_TILES (N_NODES / 16)   // 3125, exact
#define NEG_SLOPE 0.01f
#define BN_EPS 1e-5f

// ------------------------------ utility ------------------------------------
__global__ void fill_f32(float* __restrict__ p, float v, int n) {
  int i = blockIdx.x * blockDim.x + threadIdx.x;
  if (i < n) p[i] = v;
}

// W (64 x N, row major) -> WT (N x 64, row major)
__global__ void transpose_w(const float* __restrict__ W, float* __restrict__ WT, int N) {
  int idx = blockIdx.x * blockDim.x + threadIdx.x;
  if (idx < 64 * N) {
    int k = idx / N, n = idx - k * N;
    WT[(size_t)n * 64 + k] = W[idx];
  }
}

// deg[d] += 1 per edge (deg pre-initialized to 1.0 for the self loop)
__global__ void deg_count(const int* __restrict__ dst, float* __restrict__ deg, int E) {
  int e = blockIdx.x * blockDim.x + threadIdx.x;
  if (e < E) atomicAdd(&deg[dst[e]], 1.0f);
}

__global__ void compute_dinv(const float* __restrict__ deg, float* __restrict__ dinv, int n) {
  int i = blockIdx.x * blockDim.x + threadIdx.x;
  if (i < n) {
    float d = deg[i];
    dinv[i] = d > 0.0f ? __frsqrt_rn(d) : 0.0f;
  }
}

__global__ void compute_norm(const int* __restrict__ src, const int* __restrict__ dst,
                             const float* __restrict__ dinv, float* __restrict__ norm, int E) {
  int e = blockIdx.x * blockDim.x + threadIdx.x;
  if (e < E) norm[e] = dinv[src[e]] * dinv[dst[e]];
}

// ------------------------- aggregation (A * H) ------------------------------
// self-loop init, width 3: out[i][c] = dinv[i]^2 * h[i][c]
__global__ void agg_self_init3(const float* __restrict__ h, const float* __restrict__ dinv,
                               float* __restrict__ out, int n) {
  int idx = blockIdx.x * blockDim.x + threadIdx.x;
  if (idx < n * 3) {
    int i = idx / 3;
    float w = dinv[i];
    out[idx] = w * w * h[idx];
  }
}

// self-loop init, width 64 (float4 vectorized)
__global__ void agg_self_init64(const float* __restrict__ h, const float* __restrict__ dinv,
                                float* __restrict__ out, int n) {
  int idx = blockIdx.x * blockDim.x + threadIdx.x;   // over n*16 float4s
  if (idx < n * 16) {
    int i = idx >> 4;
    float w = dinv[i];
    float w2 = w * w;
    float4 v = ((const float4*)h)[idx];
    v.x *= w2; v.y *= w2; v.z *= w2; v.w *= w2;
    ((float4*)out)[idx] = v;
  }
}

// edge scatter: out[dst][c] += norm[e] * h[src][c]   (one thread per (e, c))
__global__ void agg_edges64(const float* __restrict__ h, const int* __restrict__ src,
                            const int* __restrict__ dst, const float* __restrict__ norm,
                            float* __restrict__ out, int E) {
  long long t = (long long)blockIdx.x * blockDim.x + threadIdx.x;
  int c = (int)(t & 63);
  int e = (int)(t >> 6);
  if (e < E) {
    int s = src[e], d = dst[e];
    float w = norm[e];
    atomicAdd(&out[(size_t)d * 64 + c], w * h[(size_t)s * 64 + c]);
  }
}

__global__ void agg_edges3(const float* __restrict__ h, const int* __restrict__ src,
                           const int* __restrict__ dst, const float* __restrict__ norm,
                           float* __restrict__ out, int E) {
  int e = blockIdx.x * blockDim.x + threadIdx.x;
  if (e < E) {
    int s = src[e], d = dst[e];
    float w = norm[e];
#pragma unroll
    for (int c = 0; c < 3; ++c)
      atomicAdd(&out[(size_t)d * 3 + c], w * h[(size_t)s * 3 + c]);
  }
}

// ----------------------- layer 1: K=3 scalar GEMM ---------------------------
// 4 output channels per thread; K=3 is too thin for WMMA.
__global__ void gemm3_leaky(const float* __restrict__ A, const float* __restrict__ W,
                            const float* __restrict__ b, float* __restrict__ out, int n) {
  int idx = blockIdx.x * blockDim.x + threadIdx.x;   // over n*16
  if (idx < n * 16) {
    int i = idx >> 4;
    int q = idx & 15;            // channel group
    const float* a = A + (size_t)i * 3;
    float a0 = a[0], a1 = a[1], a2 = a[2];
    float4 w0 = ((const float4*)(W))[q];          // W[0][4q..4q+3]
    float4 w1 = ((const float4*)(W + 64))[q];     // W[1][..]
    float4 w2 = ((const float4*)(W + 128))[q];    // W[2][..]
    float4 bv = ((const float4*)b)[q];
    float4 r;
    r.x = fmaf(a0, w0.x, fmaf(a1, w1.x, fmaf(a2, w2.x, bv.x)));
    r.y = fmaf(a0, w0.y, fmaf(a1, w1.y, fmaf(a2, w2.y, bv.y)));
    r.z = fmaf(a0, w0.z, fmaf(a1, w1.z, fmaf(a2, w2.z, bv.z)));
    r.w = fmaf(a0, w0.w, fmaf(a1, w1.w, fmaf(a2, w2.w, bv.w)));
    r.x = r.x >= 0.0f ? r.x : NEG_SLOPE * r.x;
    r.y = r.y >= 0.0f ? r.y : NEG_SLOPE * r.y;
    r.z = r.z >= 0.0f ? r.z : NEG_SLOPE * r.z;
    r.w = r.w >= 0.0f ? r.w : NEG_SLOPE * r.w;
    ((float4*)out)[idx] = r;
  }
}

// ----------------------- WMMA GEMM: M x 64 @ 64 x N -------------------------
// One wave computes a 16x64 output tile: 4 C-fragments, A-fragment reused.
// A: row-major Mx64.  WT: row-major Nx64 (pre-transposed weight).
// ISA layouts (CDNA5 §7.12.2):
//   A 16x4 f32:  lane&15 = row M, (lane>>4)*2 = K base, v2f = {K, K+1}
//   B 4x16 f32:  lane&15 = col N, (lane>>4)*2 = K base, v2f = {K, K+1}
//   C/D 16x16:   lane&15 = col N, vgpr r = row (lane>=16 -> row r+8)
template <bool LEAKY>
__global__ __launch_bounds__(128) void gemm_wmma(const float* __restrict__ A,
                                                 const float* __restrict__ WT,
                                                 const float* __restrict__ bias,
                                                 float* __restrict__ out,
                                                 int mtiles, int N) {
  int tileM = blockIdx.x * blockDim.y + threadIdx.y;   // wave-uniform
  if (tileM >= mtiles) return;                         // uniform branch: EXEC stays full
  int colBase = blockIdx.y * 64;
  int lane = threadIdx.x;
  int kh = lane >> 4;        // K-half select (0/1)
  int l15 = lane & 15;

  const float* arow = A + ((size_t)(tileM * 16 + l15)) * 64 + kh * 2;
  const float* wrow = WT + ((size_t)(colBase + l15)) * 64 + kh * 2;

  v8f c0 = {}, c1 = {}, c2 = {}, c3 = {};
  for (int k0 = 0; k0 < 64; k0 += 4) {
    v2f a  = *(const v2f*)(arow + k0);
    v2f b0 = *(const v2f*)(wrow + k0);
    v2f b1 = *(const v2f*)(wrow + 16 * 64 + k0);
    v2f b2 = *(const v2f*)(wrow + 32 * 64 + k0);
    v2f b3 = *(const v2f*)(wrow + 48 * 64 + k0);
    c0 = __builtin_amdgcn_wmma_f32_16x16x4_f32(false, a, false, b0, (short)0, c0, false, false);
    c1 = __builtin_amdgcn_wmma_f32_16x16x4_f32(false, a, false, b1, (short)0, c1, false, false);
    c2 = __builtin_amdgcn_wmma_f32_16x16x4_f32(false, a, false, b2, (short)0, c2, false, false);
    c3 = __builtin_amdgcn_wmma_f32_16x16x4_f32(false, a, false, b3, (short)0, c3, false, false);
  }

  v8f acc[4] = {c0, c1, c2, c3};
  size_t rowBase = (size_t)(tileM * 16 + kh * 8);
#pragma unroll
  for (int j = 0; j < 4; ++j) {
    int col = colBase + j * 16 + l15;
    float bv = bias[col];
    float* oc = out + rowBase * N + col;
#pragma unroll
    for (int r = 0; r < 8; ++r) {
      float v = acc[j][r] + bv;
      if (LEAKY) v = v >= 0.0f ? v : NEG_SLOPE * v;
      oc[(size_t)r * N] = v;
    }
  }
}

// --------------------------- BatchNorm (train stats) ------------------------
// float4 loads: each thread accumulates 4 adjacent channels over 1/16 of rows.
__global__ void colstats(const float* __restrict__ h, float* __restrict__ colsum,
                         float* __restrict__ colsumsq, int M) {
  __shared__ float ss[256][4];
  __shared__ float qq[256][4];
  int tid = threadIdx.x;
  int cg = tid & 15;   // column group (4 channels)
  int rs = tid >> 4;   // row stream 0..15
  float s0 = 0, s1 = 0, s2 = 0, s3 = 0;
  float q0 = 0, q1 = 0, q2 = 0, q3 = 0;
  const float4* h4 = (const float4*)h;
  for (int r = blockIdx.x * 16 + rs; r < M; r += gridDim.x * 16) {
    float4 v = h4[(size_t)r * 16 + cg];
    s0 += v.x; q0 += v.x * v.x;
    s1 += v.y; q1 += v.y * v.y;
    s2 += v.z; q2 += v.z * v.z;
    s3 += v.w; q3 += v.w * v.w;
  }
  ss[tid][0] = s0; ss[tid][1] = s1; ss[tid][2] = s2; ss[tid][3] = s3;
  qq[tid][0] = q0; qq[tid][1] = q1; qq[tid][2] = q2; qq[tid][3] = q3;
  __syncthreads();
  if (tid < 16) {
    float a0 = 0, a1 = 0, a2 = 0, a3 = 0;
    float b0 = 0, b1 = 0, b2 = 0, b3 = 0;
#pragma unroll
    for (int r = 0; r < 16; ++r) {
      int t = r * 16 + tid;
      a0 += ss[t][0]; a1 += ss[t][1]; a2 += ss[t][2]; a3 += ss[t][3];
      b0 += qq[t][0]; b1 += qq[t][1]; b2 += qq[t][2]; b3 += qq[t][3];
    }
    atomicAdd(&colsum[tid * 4 + 0], a0);
    atomicAdd(&colsum[tid * 4 + 1], a1);
    atomicAdd(&colsum[tid * 4 + 2], a2);
    atomicAdd(&colsum[tid * 4 + 3], a3);
    atomicAdd(&colsumsq[tid * 4 + 0], b0);
    atomicAdd(&colsumsq[tid * 4 + 1], b1);
    atomicAdd(&colsumsq[tid * 4 + 2], b2);
    atomicAdd(&colsumsq[tid * 4 + 3], b3);
  }
}

__global__ void bn_finalize(const float* __restrict__ colsum, const float* __restrict__ colsumsq,
                            const float* __restrict__ gamma, const float* __restrict__ beta,
                            float* __restrict__ scale, float* __restrict__ shift, float Mf) {
  int c = threadIdx.x;
  if (c < 64) {
    float mean = colsum[c] / Mf;
    float var = colsumsq[c] / Mf - mean * mean;
    var = var > 0.0f ? var : 0.0f;
    float inv = __frsqrt_rn(var + BN_EPS);
    float sc = gamma[c] * inv;
    scale[c] = sc;
    shift[c] = beta[c] - mean * sc;
  }
}

__global__ void bn_apply(float* __restrict__ h, const float* __restrict__ scale,
                         const float* __restrict__ shift, int M) {
  int idx = blockIdx.x * blockDim.x + threadIdx.x;   // over M*16 float4s
  if (idx < M * 16) {
    int q = idx & 15;
    int c0 = q * 4;
    float4 v = ((const float4*)h)[idx];
    v.x = fmaf(v.x, scale[c0 + 0], shift[c0 + 0]);
    v.y = fmaf(v.y, scale[c0 + 1], shift[c0 + 1]);
    v.z = fmaf(v.z, scale[c0 + 2], shift[c0 + 2]);
    v.w = fmaf(v.w, scale[c0 + 3], shift[c0 + 3]);
    ((float4*)h)[idx] = v;
  }
}

// ------------------------------- launcher -----------------------------------
extern "C" void kernel_launch(void* const* d_in, const int* in_sizes, int n_in,
                              void* d_out, int out_size, void* d_ws, size_t ws_size,
                              hipStream_t stream) {
  const float* x   = (const float*)d_in[0];
  const int*   ei  = (const int*)d_in[1];   // jnp "int64" is int32 (x64 disabled)
  const float* W1  = (const float*)d_in[2];
  const float* b1  = (const float*)d_in[3];
  const float* g1  = (const float*)d_in[4];
  const float* be1 = (const float*)d_in[5];
  const float* W2  = (const float*)d_in[6];
  const float* b2  = (const float*)d_in[7];
  const float* g2  = (const float*)d_in[8];
  const float* be2 = (const float*)d_in[9];
  const float* W3  = (const float*)d_in[10];
  const float* b3  = (const float*)d_in[11];
  const float* g3  = (const float*)d_in[12];
  const float* be3 = (const float*)d_in[13];
  const float* Wmu = (const float*)d_in[14];
  const float* bmu = (const float*)d_in[15];
  const float* Wls = (const float*)d_in[16];
  const float* bls = (const float*)d_in[17];
  float* out = (float*)d_out;

  const int N = N_NODES, E = N_EDGES;
  const int* src = ei;
  const int* dst = ei + E;

  // workspace carve (floats)
  float* ws = (float*)d_ws;
  float* deg   = ws;                 ws += N;
  float* dinv  = ws;                 ws += N;
  float* norm  = ws;                 ws += E;
  float* agg1  = ws;                 ws += (size_t)N * 3 + 4;  // keep next 16B-aligned
  float* hA    = ws;                 ws += (size_t)N * HID;
  float* hB    = ws;                 ws += (size_t)N * HID;
  float* aggH  = ws;                 ws += (size_t)N * HID;
  float* WT    = ws;                 ws += (size_t)OUTC * 64;  // transposed weight (max size)
  float* colsum   = ws;              ws += 64;
  float* colsumsq = ws;              ws += 64;
  float* bnscale  = ws;              ws += 64;
  float* bnshift  = ws;              ws += 64;

  const int T = 256;
  dim3 wblk(32, 4);  // 4 waves / block
  dim3 wg64((M_TILES + 3) / 4, 1);
  dim3 wg768((M_TILES + 3) / 4, OUTC / 64);

  // degrees / edge norms (shared across all layers)
  hipLaunchKernelGGL(fill_f32, dim3((N + T - 1) / T), dim3(T), 0, stream, deg, 1.0f, N);
  hipLaunchKernelGGL(deg_count, dim3((E + T - 1) / T), dim3(T), 0, stream, dst, deg, E);
  hipLaunchKernelGGL(compute_dinv, dim3((N + T - 1) / T), dim3(T), 0, stream, deg, dinv, N);
  hipLaunchKernelGGL(compute_norm, dim3((E + T - 1) / T), dim3(T), 0, stream, src, dst, dinv, norm, E);

  long long e64 = (long long)E * 64;
  int eblk64 = (int)((e64 + T - 1) / T);
  int nv4 = N * 16;  // float4 count for N x 64

  // ---- block 1: agg(x) @ W1, width 3 ----
  hipLaunchKernelGGL(agg_self_init3, dim3((N * 3 + T - 1) / T), dim3(T), 0, stream, x, dinv, agg1, N);
  hipLaunchKernelGGL(agg_edges3, dim3((E + T - 1) / T), dim3(T), 0, stream, x, src, dst, norm, agg1, E);
  hipLaunchKernelGGL(gemm3_leaky, dim3((nv4 + T - 1) / T), dim3(T), 0, stream, agg1, W1, b1, hA, N);
  hipLaunchKernelGGL(fill_f32, dim3(1), dim3(T), 0, stream, colsum, 0.0f, 128);  // colsum+colsumsq
  hipLaunchKernelGGL(colstats, dim3(128), dim3(T), 0, stream, hA, colsum, colsumsq, N);
  hipLaunchKernelGGL(bn_finalize, dim3(1), dim3(64), 0, stream, colsum, colsumsq, g1, be1, bnscale, bnshift, (float)N);
  hipLaunchKernelGGL(bn_apply, dim3((nv4 + T - 1) / T), dim3(T), 0, stream, hA, bnscale, bnshift, N);

  // ---- block 2 ----
  hipLaunchKernelGGL(agg_self_init64, dim3((nv4 + T - 1) / T), dim3(T), 0, stream, hA, dinv, aggH, N);
  hipLaunchKernelGGL(agg_edges64, dim3(eblk64), dim3(T), 0, stream, hA, src, dst, norm, aggH, E);
  hipLaunchKernelGGL(transpose_w, dim3((64 * HID + T - 1) / T), dim3(T), 0, stream, W2, WT, HID);
  hipLaunchKernelGGL((gemm_wmma<true>), wg64, wblk, 0, stream, aggH, WT, b2, hB, M_TILES, HID);
  hipLaunchKernelGGL(fill_f32, dim3(1), dim3(T), 0, stream, colsum, 0.0f, 128);
  hipLaunchKernelGGL(colstats, dim3(128), dim3(T), 0, stream, hB, colsum, colsumsq, N);
  hipLaunchKernelGGL(bn_finalize, dim3(1), dim3(64), 0, stream, colsum, colsumsq, g2, be2, bnscale, bnshift, (float)N);
  hipLaunchKernelGGL(bn_apply, dim3((nv4 + T - 1) / T), dim3(T), 0, stream, hB, bnscale, bnshift, N);

  // ---- block 3 ----
  hipLaunchKernelGGL(agg_self_init64, dim3((nv4 + T - 1) / T), dim3(T), 0, stream, hB, dinv, aggH, N);
  hipLaunchKernelGGL(agg_edges64, dim3(eblk64), dim3(T), 0, stream, hB, src, dst, norm, aggH, E);
  hipLaunchKernelGGL(transpose_w, dim3((64 * HID + T - 1) / T), dim3(T), 0, stream, W3, WT, HID);
  hipLaunchKernelGGL((gemm_wmma<true>), wg64, wblk, 0, stream, aggH, WT, b3, hA, M_TILES, HID);
  hipLaunchKernelGGL(fill_f32, dim3(1), dim3(T), 0, stream, colsum, 0.0f, 128);
  hipLaunchKernelGGL(colstats, dim3(128), dim3(T), 0, stream, hA, colsum, colsumsq, N);
  hipLaunchKernelGGL(bn_finalize, dim3(1), dim3(64), 0, stream, colsum, colsumsq, g3, be3, bnscale, bnshift, (float)N);
  hipLaunchKernelGGL(bn_apply, dim3((nv4 + T - 1) / T), dim3(T), 0, stream, hA, bnscale, bnshift, N);

  // ---- heads: ONE shared width-64 aggregation, two 768-wide WMMA GEMMs ----
  hipLaunchKernelGGL(agg_self_init64, dim3((nv4 + T - 1) / T), dim3(T), 0, stream, hA, dinv, aggH, N);
  hipLaunchKernelGGL(agg_edges64, dim3(eblk64), dim3(T), 0, stream, hA, src, dst, norm, aggH, E);
  hipLaunchKernelGGL(transpose_w, dim3((64 * OUTC + T - 1) / T), dim3(T), 0, stream, Wmu, WT, OUTC);
  hipLaunchKernelGGL((gemm_wmma<false>), wg768, wblk, 0, stream, aggH, WT, bmu, out, M_TILES, OUTC);
  hipLaunchKernelGGL(transpose_w, dim3((64 * OUTC + T - 1) / T), dim3(T), 0, stream, Wls, WT, OUTC);
  hipLaunchKernelGGL((gemm_wmma<false>), wg768, wblk, 0, stream, aggH, WT, bls,
                     out + (size_t)N * OUTC, M_TILES, OUTC);

  (void)in_sizes; (void)n_in; (void)out_size; (void)ws_size;
}